// MoE_46334107189619
// MI455X (gfx1250) — compile-verified
//
#include <hip/hip_runtime.h>
#include <math.h>

// ---------------------------------------------------------------------------
// MoE (E=32, D=768, F=2048, T=2048, top-2) for gfx1250 (MI455X).
// Memory-bound problem: 604MB of expert weights @ 23.3 TB/s ~= 26us floor.
// Grouped-GEMM over *selected* experts only (16x less compute than dense scan),
// fp32 WMMA (v_wmma_f32_16x16x4_f32) to keep reference fp32 numerics.
// ---------------------------------------------------------------------------

typedef __attribute__((ext_vector_type(2))) float v2f;
typedef __attribute__((ext_vector_type(8))) float v8f;

constexpr int E_ = 32;     // experts
constexpr int D_ = 768;    // embed dim
constexpr int F_ = 2048;   // ff dim
constexpr int T_ = 2048;   // tokens (B*S)
constexpr int K_ = 2;      // top_k
constexpr int SEL_ = T_ * K_;  // 4096 selections; slot s = t*2 + k

// ---------------------------------------------------------------------------
// Router: 1 wave per token, lane = expert. logits -> softmax -> top-2.
// ---------------------------------------------------------------------------
__global__ __launch_bounds__(256) void moe_router(
    const float* __restrict__ x, const float* __restrict__ rw,
    int* __restrict__ top_i, float* __restrict__ top_v) {
  const int lane = threadIdx.x & 31;
  const int wave = threadIdx.x >> 5;
  const int t = blockIdx.x * 8 + wave;
  if (t >= T_) return;

  const float* xr = x + (size_t)t * D_;
  float acc = 0.f;
#pragma unroll 4
  for (int d = 0; d < D_; ++d) acc = fmaf(xr[d], rw[d * E_ + lane], acc);

  // softmax over the 32 lanes (wave32)
  float mx = acc;
  for (int off = 16; off; off >>= 1) mx = fmaxf(mx, __shfl_xor(mx, off, 32));
  float ex = __expf(acc - mx);
  float sum = ex;
  for (int off = 16; off; off >>= 1) sum += __shfl_xor(sum, off, 32);
  const float p = ex / sum;

  // top-1
  float v1 = p; int i1 = lane;
  for (int off = 16; off; off >>= 1) {
    float ov = __shfl_xor(v1, off, 32);
    int   oi = __shfl_xor(i1, off, 32);
    if (ov > v1 || (ov == v1 && oi < i1)) { v1 = ov; i1 = oi; }
  }
  // top-2 (mask out top-1 lane)
  float v2 = (lane == i1) ? -1.f : p; int i2 = lane;
  for (int off = 16; off; off >>= 1) {
    float ov = __shfl_xor(v2, off, 32);
    int   oi = __shfl_xor(i2, off, 32);
    if (ov > v2 || (ov == v2 && oi < i2)) { v2 = ov; i2 = oi; }
  }
  if (lane == 0) {
    top_i[t * 2 + 0] = i1; top_v[t * 2 + 0] = v1;
    top_i[t * 2 + 1] = i2; top_v[t * 2 + 1] = v2;
  }
}

// ---------------------------------------------------------------------------
// Bucket selections by expert. List order is nondeterministic but harmless:
// each selection's math is row-independent and lands at its fixed slot.
// ---------------------------------------------------------------------------
__global__ void moe_zero(int* __restrict__ cnt) {
  if (threadIdx.x < E_) cnt[threadIdx.x] = 0;
}

__global__ __launch_bounds__(256) void moe_assign(
    const int* __restrict__ top_i, int* __restrict__ cnt,
    int* __restrict__ list_sel) {
  const int s = blockIdx.x * blockDim.x + threadIdx.x;
  if (s >= SEL_) return;
  const int e = top_i[s];
  const int pos = atomicAdd(&cnt[e], 1);
  list_sel[e * T_ + pos] = s;
}

// ---------------------------------------------------------------------------
// Gate/Up grouped GEMM + fused SiLU.
// Block = (token-tile of 16, expert). 8 waves split F into 16-wide N-tiles.
// A tile (16 gathered x rows) staged in LDS; padded stride 772 (bank-safe).
// ---------------------------------------------------------------------------
constexpr int AST = D_ + 4;  // 772: 772 % 64 == 4 -> conflict-free column reads

__global__ __launch_bounds__(256) void moe_gateup(
    const float* __restrict__ x, const float* __restrict__ gate_w,
    const float* __restrict__ up_w, const int* __restrict__ cnt,
    const int* __restrict__ list_sel, float* __restrict__ h_buf) {
  __shared__ float As[16 * AST];
  __shared__ int Slist[16];

  const int e = blockIdx.y;
  const int n_e = cnt[e];
  const int base = blockIdx.x * 16;
  if (base >= n_e) return;  // uniform early-exit (most blocks)
  const int rem = (n_e - base < 16) ? (n_e - base) : 16;

  const int tid = threadIdx.x;
  if (tid < 16) Slist[tid] = (tid < rem) ? list_sel[e * T_ + base + tid] : -1;
  __syncthreads();

  // stage gathered x rows (float4), zero-pad missing rows
  for (int idx = tid; idx < 16 * (D_ / 4); idx += 256) {
    const int r = idx / (D_ / 4);
    const int c = (idx % (D_ / 4)) * 4;
    float4 vv = make_float4(0.f, 0.f, 0.f, 0.f);
    const int sel = Slist[r];
    if (sel >= 0) vv = *(const float4*)(x + (size_t)(sel >> 1) * D_ + c);
    *(float4*)(&As[r * AST + c]) = vv;
  }
  __syncthreads();

  const int lane = tid & 31, wave = tid >> 5;
  const int mrow = lane & 15;
  const int kk = (lane >> 4) << 1;     // K sub-offset: 0 or 2
  const int mhalf = (lane >> 4) << 3;  // C row half: 0 or 8
  const float* gw = gate_w + (size_t)e * ((size_t)D_ * F_);
  const float* uw = up_w + (size_t)e * ((size_t)D_ * F_);

  for (int nt = wave; nt < F_ / 16; nt += 8) {
    const int n0 = nt * 16;
    v8f accg = {0, 0, 0, 0, 0, 0, 0, 0};
    v8f accu = {0, 0, 0, 0, 0, 0, 0, 0};
    const float* gp = gw + n0 + mrow;
    const float* up = uw + n0 + mrow;
    __builtin_prefetch(gp, 0, 1);  // global_prefetch_b8 on the weight stream
    __builtin_prefetch(up, 0, 1);
#pragma unroll 4
    for (int k0 = 0; k0 < D_; k0 += 4) {
      const v2f a = *(const v2f*)(&As[mrow * AST + k0 + kk]);
      const size_t ro = (size_t)(k0 + kk) * F_;
      v2f bg, bu;
      bg.x = gp[ro]; bg.y = gp[ro + F_];
      bu.x = up[ro]; bu.y = up[ro + F_];
      accg = __builtin_amdgcn_wmma_f32_16x16x4_f32(false, a, false, bg,
                                                   (short)0, accg, false, false);
      accu = __builtin_amdgcn_wmma_f32_16x16x4_f32(false, a, false, bu,
                                                   (short)0, accu, false, false);
    }
    // h = silu(g) * u ; scatter rows to their selection slots
#pragma unroll
    for (int r = 0; r < 8; ++r) {
      const int m = r + mhalf;
      if (m < rem) {
        const float g = accg[r], u = accu[r];
        const float hval = (g / (1.f + __expf(-g))) * u;
        h_buf[(size_t)Slist[m] * F_ + n0 + mrow] = hval;
      }
    }
  }
}

// ---------------------------------------------------------------------------
// Down grouped GEMM: y[s, :] = h[s, :] @ down_w[e]   (K=2048 chunked via LDS)
// 8 waves x 6 accumulators cover D=768 (48 N-tiles of 16).
// ---------------------------------------------------------------------------
constexpr int KC = 512;
constexpr int AST2 = KC + 4;  // 516

__global__ __launch_bounds__(256) void moe_down(
    const float* __restrict__ h_buf, const float* __restrict__ down_w,
    const int* __restrict__ cnt, const int* __restrict__ list_sel,
    float* __restrict__ y_buf) {
  __shared__ float As[16 * AST2];
  __shared__ int Slist[16];

  const int e = blockIdx.y;
  const int n_e = cnt[e];
  const int base = blockIdx.x * 16;
  if (base >= n_e) return;
  const int rem = (n_e - base < 16) ? (n_e - base) : 16;

  const int tid = threadIdx.x;
  if (tid < 16) Slist[tid] = (tid < rem) ? list_sel[e * T_ + base + tid] : -1;

  const int lane = tid & 31, wave = tid >> 5;
  const int mrow = lane & 15;
  const int kk = (lane >> 4) << 1;
  const int mhalf = (lane >> 4) << 3;
  const float* dw = down_w + (size_t)e * ((size_t)F_ * D_);

  v8f acc[6];
#pragma unroll
  for (int i = 0; i < 6; ++i) acc[i] = (v8f){0, 0, 0, 0, 0, 0, 0, 0};

  for (int kc = 0; kc < F_; kc += KC) {
    __syncthreads();  // protect As reuse across chunks (also covers Slist)
    for (int idx = tid; idx < 16 * (KC / 4); idx += 256) {
      const int r = idx / (KC / 4);
      const int c = (idx % (KC / 4)) * 4;
      float4 vv = make_float4(0.f, 0.f, 0.f, 0.f);
      const int sel = Slist[r];
      if (sel >= 0) vv = *(const float4*)(h_buf + (size_t)sel * F_ + kc + c);
      *(float4*)(&As[r * AST2 + c]) = vv;
    }
    __syncthreads();
#pragma unroll
    for (int i = 0; i < 6; ++i) {
      const int n0 = (wave + 8 * i) * 16;
      const float* dp = dw + (size_t)kc * D_ + n0 + mrow;
      __builtin_prefetch(dp, 0, 1);
      v8f a_ = acc[i];
#pragma unroll 4
      for (int k0 = 0; k0 < KC; k0 += 4) {
        const v2f a = *(const v2f*)(&As[mrow * AST2 + k0 + kk]);
        const size_t ro = (size_t)(k0 + kk) * D_;
        v2f b;
        b.x = dp[ro]; b.y = dp[ro + D_];
        a_ = __builtin_amdgcn_wmma_f32_16x16x4_f32(false, a, false, b,
                                                   (short)0, a_, false, false);
      }
      acc[i] = a_;
    }
  }
  // write per-selection rows (unweighted; combine applies router probs)
#pragma unroll
  for (int i = 0; i < 6; ++i) {
    const int n0 = (wave + 8 * i) * 16;
#pragma unroll
    for (int r = 0; r < 8; ++r) {
      const int m = r + mhalf;
      if (m < rem) y_buf[(size_t)Slist[m] * D_ + n0 + mrow] = acc[i][r];
    }
  }
}

// ---------------------------------------------------------------------------
// out[t] = w0 * y[2t] + w1 * y[2t+1]  (deterministic weighted merge, no atomics)
// ---------------------------------------------------------------------------
__global__ __launch_bounds__(256) void moe_combine(
    const float* __restrict__ y_buf, const float* __restrict__ top_v,
    float* __restrict__ out) {
  const int i = blockIdx.x * blockDim.x + threadIdx.x;  // over T_*D_/4
  if (i >= T_ * (D_ / 4)) return;
  const int t = i / (D_ / 4);
  const int c = (i % (D_ / 4)) * 4;
  const float w0 = top_v[t * 2], w1 = top_v[t * 2 + 1];
  const float4 a = *(const float4*)(y_buf + (size_t)(t * 2) * D_ + c);
  const float4 b = *(const float4*)(y_buf + (size_t)(t * 2 + 1) * D_ + c);
  float4 o;
  o.x = w0 * a.x + w1 * b.x;
  o.y = w0 * a.y + w1 * b.y;
  o.z = w0 * a.z + w1 * b.z;
  o.w = w0 * a.w + w1 * b.w;
  *(float4*)(out + (size_t)t * D_ + c) = o;
}

// ---------------------------------------------------------------------------
extern "C" void kernel_launch(void* const* d_in, const int* in_sizes, int n_in,
                              void* d_out, int out_size, void* d_ws,
                              size_t ws_size, hipStream_t stream) {
  const float* x        = (const float*)d_in[0];  // [T, D]
  const float* router_w = (const float*)d_in[1];  // [D, E]
  const float* gate_w   = (const float*)d_in[2];  // [E, D, F]
  const float* up_w     = (const float*)d_in[3];  // [E, D, F]
  const float* down_w   = (const float*)d_in[4];  // [E, F, D]
  float* out = (float*)d_out;                     // [T, D]
  (void)in_sizes; (void)n_in; (void)out_size; (void)ws_size;

  // workspace carve-up (all 128B-aligned by construction)
  char* w = (char*)d_ws;
  float* h_buf = (float*)w;  w += (size_t)SEL_ * F_ * sizeof(float);  // 33.5 MB
  float* y_buf = (float*)w;  w += (size_t)SEL_ * D_ * sizeof(float);  // 12.6 MB
  int*   top_i = (int*)w;    w += (size_t)SEL_ * sizeof(int);
  float* top_v = (float*)w;  w += (size_t)SEL_ * sizeof(float);
  int*   cnt   = (int*)w;    w += 128;
  int*   list_sel = (int*)w; w += (size_t)E_ * T_ * sizeof(int);

  moe_zero<<<1, 32, 0, stream>>>(cnt);
  moe_router<<<T_ / 8, 256, 0, stream>>>(x, router_w, top_i, top_v);
  moe_assign<<<SEL_ / 256, 256, 0, stream>>>(top_i, cnt, list_sel);
  moe_gateup<<<dim3(T_ / 16, E_), 256, 0, stream>>>(x, gate_w, up_w, cnt,
                                                    list_sel, h_buf);
  moe_down<<<dim3(T_ / 16, E_), 256, 0, stream>>>(h_buf, down_w, cnt,
                                                  list_sel, y_buf);
  moe_combine<<<(T_ * (D_ / 4) + 255) / 256, 256, 0, stream>>>(y_buf, top_v,
                                                               out);
}